// DenseWrite_63642825392619
// MI455X (gfx1250) — compile-verified
//
#include <hip/hip_runtime.h>
#include <math.h>

// Problem constants (match the reference)
#define NTOK 8192
#define KEXP 4
#define BW   16
#define DDIM 512
#define MEXP 64
#define NCOL (MEXP * BW)          // 1024 columns of Un (expert*16 + b)
#define UROWS (DDIM + 1)          // 513

typedef __attribute__((ext_vector_type(2))) float v2f;
typedef __attribute__((ext_vector_type(8))) float v8f;

// ---------------------------------------------------------------------------
// K1: normalize U over the (D+1) axis; emit two layouts:
//   Untp: pair-interleaved row-major  Untp[(d>>1)*2*NCOL + c*2 + (d&1)] = Un[c,d]
//         -> the Gram GEMM loads {Un[d][c], Un[d+1][c]} as one b64
//   Unc : column-major  Unc[c*DDIM + d] = Un[c,d]
//         -> per-token matvec streams rows with float4 coalesced loads
// One wave per column c = m*16+b; lanes cover d = lane + 32t, t<16; lane 0
// folds in the padded row d=512 (it participates in the norm only).
// ---------------------------------------------------------------------------
__global__ __launch_bounds__(128) void normalize_kernel(
    const float* __restrict__ U, float* __restrict__ Untp, float* __restrict__ Unc) {
  const int lane = threadIdx.x & 31;
  const int wave = threadIdx.x >> 5;
  const int c = blockIdx.x * 4 + wave;          // 0..1023
  const int m = c >> 4, b = c & 15;
  const float* base = U + (size_t)m * UROWS * BW + b;

  float v[16];
  float s = 0.f;
#pragma unroll
  for (int t = 0; t < 16; ++t) {
    float x = base[(lane + 32 * t) * BW];
    v[t] = x;
    s += x * x;
  }
  if (lane == 0) {                              // padded row d = 512
    float x = base[DDIM * BW];
    s += x * x;
  }
#pragma unroll
  for (int off = 16; off; off >>= 1) s += __shfl_xor(s, off, 32);
  const float inv = 1.0f / sqrtf(s);

#pragma unroll
  for (int t = 0; t < 16; ++t) {
    const int d = lane + 32 * t;
    const float x = v[t] * inv;
    Untp[(size_t)(d >> 1) * (2 * NCOL) + c * 2 + (d & 1)] = x;
    Unc[(size_t)c * DDIM + d] = x;
  }
}

// ---------------------------------------------------------------------------
// K2: Gram GEMM  G(1024x1024) = Un^T(1024x512) * Un(512x1024), K = 512,
// via V_WMMA_F32_16X16X4_F32.  Wave computes a 16x64 strip (1 A tile reused
// across 4 B tiles).  Block = 8 waves -> 128x64 of C.  Grid = (16, 8).
//
// Operand lane layouts (ISA 7.12.2, wave32):
//   A 16x4 : lanes 0-15 M=lane {K0,K1} in vgpr0/1; lanes 16-31 {K2,K3}
//   B 4x16 : vgpr v: lanes 0-15 row K=v, lanes 16-31 row K=v+2, N=lane&15
//   C 16x16: vgpr r: M=r (lanes 0-15) / M=r+8 (lanes 16-31), N=lane&15
// With the pair-interleaved Untp layout both A and B micro-panels are single
// 8-byte loads per operand register pair.
// ---------------------------------------------------------------------------
__global__ __launch_bounds__(256) void gram_kernel(
    const float* __restrict__ Untp, float* __restrict__ G) {
  const int lane = threadIdx.x & 31;
  const int wave = threadIdx.x >> 5;
  const int l    = lane & 15;
  const int half = lane >> 4;
  const int i0 = blockIdx.y * 128 + wave * 16;  // C row base
  const int j0 = blockIdx.x * 64;               // C col base

  v8f c0 = {}, c1 = {}, c2 = {}, c3 = {};

  // base pointers for this lane at k-step d=0 (dd = d + 2*half)
  const float* pa = Untp + (size_t)half * (2 * NCOL) + (i0 + l) * 2;
  const float* pb = Untp + (size_t)half * (2 * NCOL) + (j0 + l) * 2;

#pragma unroll 2
  for (int d = 0; d < DDIM; d += 4) {
    v2f a  = *(const v2f*)(pa);
    v2f b0 = *(const v2f*)(pb);
    v2f b1 = *(const v2f*)(pb + 32);
    v2f b2 = *(const v2f*)(pb + 64);
    v2f b3 = *(const v2f*)(pb + 96);
    c0 = __builtin_amdgcn_wmma_f32_16x16x4_f32(false, a, false, b0, (short)0, c0, false, false);
    c1 = __builtin_amdgcn_wmma_f32_16x16x4_f32(false, a, false, b1, (short)0, c1, false, false);
    c2 = __builtin_amdgcn_wmma_f32_16x16x4_f32(false, a, false, b2, (short)0, c2, false, false);
    c3 = __builtin_amdgcn_wmma_f32_16x16x4_f32(false, a, false, b3, (short)0, c3, false, false);
    pa += 2 * (2 * NCOL);                       // advance 4 k-rows
    pb += 2 * (2 * NCOL);
  }

#pragma unroll
  for (int r = 0; r < 8; ++r) {
    const int row = i0 + r + 8 * half;
    float* g = G + (size_t)row * NCOL + j0 + l;
    g[0]  = c0[r];
    g[16] = c1[r];
    g[32] = c2[r];
    g[48] = c3[r];
  }
}

// ---------------------------------------------------------------------------
// K3: writes[n,d] = sum_{k,b} Un[e_nk, d, b] * h[n,k,b].
// One wave per token.  h_cat (64 floats) lives in 2 regs/lane, broadcast with
// wave32 shuffles.  Unc rows are streamed as float4 (b128, fully coalesced).
// Lane owns d = 4*lane + 128*j, j<4.
// ---------------------------------------------------------------------------
__global__ __launch_bounds__(256) void writes_kernel(
    const float* __restrict__ h, const int* __restrict__ idx,
    const float* __restrict__ Unc, float* __restrict__ out) {
  const int lane = threadIdx.x & 31;
  const int wave = threadIdx.x >> 5;
  const int n = blockIdx.x * 8 + wave;

  const float hA = h[(size_t)n * 64 + lane];
  const float hB = h[(size_t)n * 64 + 32 + lane];

  float a0x=0,a0y=0,a0z=0,a0w=0, a1x=0,a1y=0,a1z=0,a1w=0;
  float a2x=0,a2y=0,a2z=0,a2w=0, a3x=0,a3y=0,a3z=0,a3w=0;

  for (int k = 0; k < KEXP; ++k) {
    const int e = idx[n * KEXP + k];
    const float4* Ub = (const float4*)(Unc + (size_t)e * BW * DDIM);
#pragma unroll
    for (int b = 0; b < BW; ++b) {
      const int o = k * BW + b;
      const float w = (o < 32) ? __shfl(hA, o, 32) : __shfl(hB, o - 32, 32);
      const float4* row = Ub + b * (DDIM / 4);
      float4 r0 = row[lane], r1 = row[lane + 32], r2 = row[lane + 64], r3 = row[lane + 96];
      a0x = fmaf(w, r0.x, a0x); a0y = fmaf(w, r0.y, a0y); a0z = fmaf(w, r0.z, a0z); a0w = fmaf(w, r0.w, a0w);
      a1x = fmaf(w, r1.x, a1x); a1y = fmaf(w, r1.y, a1y); a1z = fmaf(w, r1.z, a1z); a1w = fmaf(w, r1.w, a1w);
      a2x = fmaf(w, r2.x, a2x); a2y = fmaf(w, r2.y, a2y); a2z = fmaf(w, r2.z, a2z); a2w = fmaf(w, r2.w, a2w);
      a3x = fmaf(w, r3.x, a3x); a3y = fmaf(w, r3.y, a3y); a3z = fmaf(w, r3.z, a3z); a3w = fmaf(w, r3.w, a3w);
    }
  }
  float4* op = (float4*)(out + (size_t)n * DDIM);
  op[lane]      = make_float4(a0x, a0y, a0z, a0w);
  op[lane + 32] = make_float4(a1x, a1y, a1z, a1w);
  op[lane + 64] = make_float4(a2x, a2y, a2z, a2w);
  op[lane + 96] = make_float4(a3x, a3y, a3z, a3w);
}

// ---------------------------------------------------------------------------
// K4: loss via Gram:  h_recon_cat = G_sub(64x64) * h_cat(64) per token, then
// sum of squared diffs.  Wave/token; lane owns outputs o=lane and o=lane+32.
// Deterministic: shuffle + LDS tree reduction -> per-block partial sums.
// ---------------------------------------------------------------------------
__global__ __launch_bounds__(256) void recon_loss_kernel(
    const float* __restrict__ h, const int* __restrict__ idx,
    const float* __restrict__ G, float* __restrict__ partials) {
  __shared__ float red[8];
  const int lane = threadIdx.x & 31;
  const int wave = threadIdx.x >> 5;
  const int n = blockIdx.x * 8 + wave;

  const int e0 = idx[n * 4 + 0], e1 = idx[n * 4 + 1];
  const int e2 = idx[n * 4 + 2], e3 = idx[n * 4 + 3];

  const float hA = h[(size_t)n * 64 + lane];
  const float hB = h[(size_t)n * 64 + 32 + lane];

  const int row0 = ((lane < 16) ? e0 : e1) * BW + (lane & 15);   // output o = lane
  const int row1 = ((lane < 16) ? e2 : e3) * BW + (lane & 15);   // output o = lane+32
  const float* g0 = G + (size_t)row0 * NCOL;
  const float* g1 = G + (size_t)row1 * NCOL;

  float acc0 = 0.f, acc1 = 0.f;
#pragma unroll
  for (int kp = 0; kp < 4; ++kp) {
    const int ek = (kp == 0) ? e0 : (kp == 1) ? e1 : (kp == 2) ? e2 : e3;
    const int colb = ek * BW;
#pragma unroll
    for (int bp = 0; bp < BW; ++bp) {
      const int op = kp * BW + bp;                                // uniform
      const float coef = (op < 32) ? __shfl(hA, op, 32) : __shfl(hB, op - 32, 32);
      acc0 = fmaf(g0[colb + bp], coef, acc0);
      acc1 = fmaf(g1[colb + bp], coef, acc1);
    }
  }
  const float d0 = acc0 - hA;
  const float d1 = acc1 - hB;
  float s = d0 * d0 + d1 * d1;
#pragma unroll
  for (int off = 16; off; off >>= 1) s += __shfl_xor(s, off, 32);
  if (lane == 0) red[wave] = s;
  __syncthreads();
  if (threadIdx.x == 0) {
    float t = 0.f;
#pragma unroll
    for (int i = 0; i < 8; ++i) t += red[i];
    partials[blockIdx.x] = t;
  }
}

// K5: final deterministic reduction of 1024 partials -> mean -> d_out[N*D]
__global__ __launch_bounds__(256) void finalize_loss_kernel(
    const float* __restrict__ partials, float* __restrict__ out_loss) {
  __shared__ float red[256];
  float s = 0.f;
  for (int i = threadIdx.x; i < 1024; i += 256) s += partials[i];
  red[threadIdx.x] = s;
  __syncthreads();
  for (int off = 128; off; off >>= 1) {
    if (threadIdx.x < off) red[threadIdx.x] += red[threadIdx.x + off];
    __syncthreads();
  }
  if (threadIdx.x == 0)
    out_loss[0] = red[0] * (1.0f / (float)(NTOK * KEXP * BW));
}

extern "C" void kernel_launch(void* const* d_in, const int* in_sizes, int n_in,
                              void* d_out, int out_size, void* d_ws, size_t ws_size,
                              hipStream_t stream) {
  const float* h   = (const float*)d_in[0];   // (N, K, B) f32
  const int*   idx = (const int*)  d_in[1];   // (N, K)    i32
  const float* U   = (const float*)d_in[2];   // (M, D+1, B) f32
  float* out = (float*)d_out;                 // writes (N*D) ++ loss (1)

  // workspace layout (floats): Untp 512K | Unc 512K | G 1M | partials 1K  (~8 MB)
  float* ws = (float*)d_ws;
  float* Untp = ws;
  float* Unc  = ws + (size_t)DDIM * NCOL;
  float* G    = ws + (size_t)2 * DDIM * NCOL;
  float* part = G + (size_t)NCOL * NCOL;

  normalize_kernel<<<dim3(NCOL / 4), dim3(128), 0, stream>>>(U, Untp, Unc);
  gram_kernel<<<dim3(NCOL / 64, NCOL / 128), dim3(256), 0, stream>>>(Untp, G);
  writes_kernel<<<dim3(NTOK / 8), dim3(256), 0, stream>>>(h, idx, Unc, out);
  recon_loss_kernel<<<dim3(NTOK / 8), dim3(256), 0, stream>>>(h, idx, G, part);
  finalize_loss_kernel<<<dim3(1), dim3(256), 0, stream>>>(part, out + (size_t)NTOK * DDIM);
}